// RBFKernel_65481071396729
// MI455X (gfx1250) — compile-verified
//
#include <hip/hip_runtime.h>

typedef __attribute__((ext_vector_type(2))) float v2f;
typedef __attribute__((ext_vector_type(4))) float v4f;
typedef __attribute__((ext_vector_type(8))) float v8f;

#define NROWS 8192
#define MROWS 8192
#define DDIM  64

#define LOG2E 1.44269504088896340736f

// ---------------------------------------------------------------------------
// Kernel 1: scaled row squared-norms, stored directly in exp2 domain:
//   bx[i] = -gamma*log2(e)*||x_i||^2 ,  by[j] = -gamma*log2(e)*||y_j||^2
// so the main-kernel epilogue is a single fma into v_exp_f32.
// ---------------------------------------------------------------------------
__global__ __launch_bounds__(256) void rbf_rownorms(
    const float* __restrict__ x, const float* __restrict__ y,
    const float* __restrict__ gamma_p,
    float* __restrict__ bx, float* __restrict__ by)
{
    int r = blockIdx.x * blockDim.x + threadIdx.x;   // 0 .. 16383
    const float* src;
    float* dst;
    int row;
    if (r < NROWS) { src = x; dst = bx; row = r; }
    else           { src = y; dst = by; row = r - NROWS; }

    const v4f* p = (const v4f*)(src + (long)row * DDIM);
    float s = 0.0f;
#pragma unroll
    for (int i = 0; i < DDIM / 4; ++i) {
        v4f v = p[i];
        s += v.x * v.x + v.y * v.y + v.z * v.z + v.w * v.w;
    }
    dst[row] = -gamma_p[0] * LOG2E * s;              // uniform gamma -> s_load
}

// ---------------------------------------------------------------------------
// Kernel 2: fused  out = exp(-gamma * max(x2 + y2 - 2*x@y^T, 0))
//                      = min( exp2( K*xy + bx + by ), 1 )   [K = 2*gamma*log2e]
// (valid since gamma >= 0 and exp is monotonic: the max-with-0 becomes a
//  min-with-1 after the exponential.)
//
// One wave32 per 32x64 output tile: 2 row-tiles x 4 col-tiles,
// 8 f32 WMMA accumulators, V_WMMA_F32_16X16X4_F32 over K=64.
//
// fp32 A (16x4) / B (4x16) fragment layout (ISA 7.12.2):
//   VGPR0: lanes 0-15 -> K=0, lanes 16-31 -> K=2
//   VGPR1: lanes 0-15 -> K=1, lanes 16-31 -> K=3
//   lane&15 = M (for A) / N (for B = y^T -> row of y): each lane loads an
//   aligned contiguous float2; all 6 fragment loads per k-step are immediate
//   offsets off two base pointers (row-tile delta = 16*64*4 = 4096 B).
//
// fp32 C/D 16x16 layout: lane -> N = lane&15 ; VGPR v -> M = v + 8*(lane>>4).
// ---------------------------------------------------------------------------
__global__ __launch_bounds__(256) void rbf_wmma_tile(
    const float* __restrict__ x, const float* __restrict__ y,
    const float* __restrict__ bxv, const float* __restrict__ byv,
    const float* __restrict__ gamma_p, float* __restrict__ out)
{
    const int lane = threadIdx.x & 31;
    const int wave = threadIdx.x >> 5;

    // 256 row-tiles (32 rows each) x 128 col-tiles (64 cols each) = 32768 waves.
    const int tIdx = blockIdx.x * 8 + wave;       // 0 .. 32767
    const int tm = tIdx >> 7;                     // 0 .. 255
    const int tn = tIdx & 127;                    // 0 .. 127
    const int i0 = tm * 32;
    const int j0 = tn * 64;

    const int mr = lane & 15;          // row-within-tile this lane gathers
    const int ks = (lane >> 4) * 2;    // K sub-offset: 0 (lanes 0-15) or 2

    const float* xp = x + (long)(i0 + mr) * DDIM + ks;
    const float* yp = y + (long)(j0 + mr) * DDIM + ks;

    v8f c[2][4];
#pragma unroll
    for (int rt = 0; rt < 2; ++rt)
#pragma unroll
        for (int ct = 0; ct < 4; ++ct)
            c[rt][ct] = (v8f){};

#pragma unroll
    for (int kk = 0; kk < DDIM; kk += 4) {
        v2f a0 = *(const v2f*)(xp + kk);
        v2f a1 = *(const v2f*)(xp + kk + 16 * DDIM);
        v2f b0 = *(const v2f*)(yp + kk);
        v2f b1 = *(const v2f*)(yp + kk + 16 * DDIM);
        v2f b2 = *(const v2f*)(yp + kk + 32 * DDIM);
        v2f b3 = *(const v2f*)(yp + kk + 48 * DDIM);
        // (neg_a, A, neg_b, B, c_mod, C, reuse_a, reuse_b)
        c[0][0] = __builtin_amdgcn_wmma_f32_16x16x4_f32(false, a0, false, b0,
                                                        (short)0, c[0][0], false, false);
        c[0][1] = __builtin_amdgcn_wmma_f32_16x16x4_f32(false, a0, false, b1,
                                                        (short)0, c[0][1], false, false);
        c[0][2] = __builtin_amdgcn_wmma_f32_16x16x4_f32(false, a0, false, b2,
                                                        (short)0, c[0][2], false, false);
        c[0][3] = __builtin_amdgcn_wmma_f32_16x16x4_f32(false, a0, false, b3,
                                                        (short)0, c[0][3], false, false);
        c[1][0] = __builtin_amdgcn_wmma_f32_16x16x4_f32(false, a1, false, b0,
                                                        (short)0, c[1][0], false, false);
        c[1][1] = __builtin_amdgcn_wmma_f32_16x16x4_f32(false, a1, false, b1,
                                                        (short)0, c[1][1], false, false);
        c[1][2] = __builtin_amdgcn_wmma_f32_16x16x4_f32(false, a1, false, b2,
                                                        (short)0, c[1][2], false, false);
        c[1][3] = __builtin_amdgcn_wmma_f32_16x16x4_f32(false, a1, false, b3,
                                                        (short)0, c[1][3], false, false);
    }

    // Epilogue: per element  min(exp2(K*c + bx + by), 1), NT streaming stores.
    const int n     = lane & 15;
    const int mhalf = (lane >> 4) * 8;
    const float K   = 2.0f * gamma_p[0] * LOG2E;  // uniform -> s_load + smul
    float by[4];
#pragma unroll
    for (int ct = 0; ct < 4; ++ct)
        by[ct] = byv[j0 + ct * 16 + n];

#pragma unroll
    for (int rt = 0; rt < 2; ++rt) {
#pragma unroll
        for (int v = 0; v < 8; ++v) {
            const int  row = i0 + rt * 16 + mhalf + v;
            const float bx = bxv[row];            // contiguous in v -> b128 loads
            float* o = out + (long)row * MROWS + j0 + n;
#pragma unroll
            for (int ct = 0; ct < 4; ++ct) {
                float arg = __builtin_fmaf(K, c[rt][ct][v], bx + by[ct]);
                float e   = __builtin_amdgcn_exp2f(arg);
                __builtin_nontemporal_store(fminf(e, 1.0f), o + ct * 16);
            }
        }
    }
}

// ---------------------------------------------------------------------------
// Launch: inputs are (x[N*D] f32, y[M*D] f32, gamma[1] f32); output f32 [N*M].
// Workspace layout: bx = ws[0:8192), by = ws[8192:16384) floats (64 KB).
// ---------------------------------------------------------------------------
extern "C" void kernel_launch(void* const* d_in, const int* in_sizes, int n_in,
                              void* d_out, int out_size, void* d_ws, size_t ws_size,
                              hipStream_t stream)
{
    const float* x = (const float*)d_in[0];
    const float* y = (const float*)d_in[1];
    const float* gamma = (const float*)d_in[2];
    float* out = (float*)d_out;

    float* bx = (float*)d_ws;
    float* by = bx + NROWS;

    // Scaled row norms: 16384 rows, 256 threads/block.
    rbf_rownorms<<<(NROWS + MROWS) / 256, 256, 0, stream>>>(x, y, gamma, bx, by);

    // Main tiles: 256 * 128 = 32768 waves, 8 waves per 256-thread block.
    rbf_wmma_tile<<<(256 * 128) / 8, 256, 0, stream>>>(x, y, bx, by, gamma, out);
}